// PairBiasAttentionLayer_79405355369238
// MI455X (gfx1250) — compile-verified
//
#include <hip/hip_runtime.h>
#include <cstdint>

// ---------------------------------------------------------------------------
// PairBiasAttention for MI455X (gfx1250), fp32 end-to-end via
// V_WMMA_F32_16X16X4_F32.
//
// Roofline: z read 256 MB (once, streaming -> NT loads) + proj_z write 128 MB
// (re-read by attention -> keep RT so it stays in the 192 MB L2) dominate;
// total FLOPs ~8.5 G are ~100x under the matrix ceiling, so fp32 WMMA keeps
// reference numerics at zero speed cost.
// ---------------------------------------------------------------------------

typedef __attribute__((ext_vector_type(2))) float v2f;
typedef __attribute__((ext_vector_type(4))) float v4f;
typedef __attribute__((ext_vector_type(8))) float v8f;

#define WMMA_F32_4(A, B, C) \
  __builtin_amdgcn_wmma_f32_16x16x4_f32(false, (A), false, (B), (short)0, (C), false, false)

constexpr int Bn = 4, Nn = 1024, Dn = 256, Hn = 8, HDn = 32, Pn = 16;
constexpr int QKVW = 3 * Dn;  // 768

// ---------------------------------------------------------------------------
// Kernel 1: LayerNorm over D=256.  One wave per row, 8 elems per lane.
// ---------------------------------------------------------------------------
__global__ void ln_x_kernel(const float* __restrict__ x, const float* __restrict__ w,
                            const float* __restrict__ bb, float* __restrict__ xn) {
  int wave = threadIdx.x >> 5;
  int lane = threadIdx.x & 31;
  int row  = blockIdx.x * 8 + wave;  // over B*N = 4096 rows
  const float* xr = x + (size_t)row * Dn;
  float v[8];
  float s = 0.f, s2 = 0.f;
#pragma unroll
  for (int i = 0; i < 8; ++i) {
    v[i] = xr[lane * 8 + i];
    s += v[i];
    s2 += v[i] * v[i];
  }
#pragma unroll
  for (int m = 1; m < 32; m <<= 1) {
    s  += __shfl_xor(s, m);
    s2 += __shfl_xor(s2, m);
  }
  float mu  = s * (1.0f / Dn);
  float var = s2 * (1.0f / Dn) - mu * mu;
  float rs  = rsqrtf(var + 1e-5f);
  float* yr = xn + (size_t)row * Dn;
#pragma unroll
  for (int i = 0; i < 8; ++i) {
    int j = lane * 8 + i;
    yr[j] = (v[i] - mu) * rs * w[j] + bb[j];
  }
}

// ---------------------------------------------------------------------------
// Kernel 2: fused QKV + gate GEMM.  x_norm[4096,256] @ [W_qkv^T | W_g^T]
// One wave computes one 16x16 output tile with 64 f32 WMMA (K chunks of 4).
// A fragment (16x4): lane 0-15 -> row=lane, K=0,1 ; lane 16-31 -> K=2,3.
// B fragment (4x16): lane 0-15 -> col=lane, K=0,1 ; lane 16-31 -> K=2,3.
// ---------------------------------------------------------------------------
__global__ void qkvg_kernel(const float* __restrict__ xn, const float* __restrict__ qkv_w,
                            const float* __restrict__ wg, const float* __restrict__ bg,
                            float* __restrict__ qkv, float* __restrict__ g) {
  int wave = threadIdx.x >> 5, lane = threadIdx.x & 31;
  int tile = blockIdx.x * 8 + wave;  // 256 row-tiles * 64 col-tiles = 16384
  int mt = tile >> 6, nt = tile & 63;
  int m0 = mt * 16, n0 = nt * 16;
  int l16 = lane & 15;
  int kh  = (lane >> 4) << 1;  // 0 or 2
  const float* arow = xn + (size_t)(m0 + l16) * Dn;
  int n = n0 + l16;
  const float* brow = (n < QKVW) ? (qkv_w + (size_t)n * Dn)
                                 : (wg + (size_t)(n - QKVW) * Dn);
  v8f c = {};
  for (int k = 0; k < Dn; k += 4) {
    v2f a, b;
    a.x = arow[k + kh];
    a.y = arow[k + kh + 1];
    b.x = brow[k + kh];
    b.y = brow[k + kh + 1];
    c = WMMA_F32_4(a, b, c);
  }
  int rbase = (lane >> 4) << 3;  // C layout: lanes 0-15 rows 0-7, 16-31 rows 8-15
  if (n < QKVW) {
#pragma unroll
    for (int v = 0; v < 8; ++v)
      qkv[(size_t)(m0 + rbase + v) * QKVW + n] = c[v];
  } else {
    float bias = bg[n - QKVW];
#pragma unroll
    for (int v = 0; v < 8; ++v) {
      float t = c[v] + bias;
      g[(size_t)(m0 + rbase + v) * Dn + (n - QKVW)] = 1.0f / (1.0f + __expf(-t));
    }
  }
}

// ---------------------------------------------------------------------------
// Kernel 3: z [B,N,N,P] -> layernorm(P) -> per-head projection -> proj_z
// output region [B,H,N,N].  One thread per (b,q,k) 16-vector.
// 256 MB read-once stream: 128-bit NT loads + global_prefetch_b8 one block
// ahead.  proj_z stores stay RT so the 128 MB lands in the 192 MB L2 for the
// attention kernel's re-read.
// ---------------------------------------------------------------------------
__global__ void projz_kernel(const float* __restrict__ z, const float* __restrict__ wln,
                             const float* __restrict__ bln, const float* __restrict__ wz,
                             const float* __restrict__ bz, float* __restrict__ out2) {
  __shared__ float s_wz[Hn * Pn];
  __shared__ float s_wln[Pn], s_bln[Pn], s_bz[Hn];
  int t = threadIdx.x;
  if (t < Hn * Pn) s_wz[t] = wz[t];
  if (t < Pn) { s_wln[t] = wln[t]; s_bln[t] = bln[t]; }
  if (t < Hn) s_bz[t] = bz[t];
  __syncthreads();

  size_t gid = (size_t)blockIdx.x * blockDim.x + t;  // over B*N*N = 4M
  const v4f* zr4 = (const v4f*)(z + gid * Pn);
  // prefetch one block-stride ahead of the stream (gfx1250 global_prefetch_b8)
  __builtin_prefetch((const float*)zr4 + (size_t)256 * Pn, 0, 0);
  v4f zq[4];
#pragma unroll
  for (int i = 0; i < 4; ++i) zq[i] = __builtin_nontemporal_load(zr4 + i);  // read-once: NT

  float zv[Pn];
  float s = 0.f, s2 = 0.f;
#pragma unroll
  for (int i = 0; i < 4; ++i) {
#pragma unroll
    for (int j = 0; j < 4; ++j) {
      float vv = zq[i][j];
      zv[i * 4 + j] = vv;
      s += vv;
      s2 += vv * vv;
    }
  }
  float mu  = s * (1.0f / Pn);
  float var = s2 * (1.0f / Pn) - mu * mu;
  float rs  = rsqrtf(var + 1e-5f);
#pragma unroll
  for (int p = 0; p < Pn; ++p) zv[p] = (zv[p] - mu) * rs * s_wln[p] + s_bln[p];

  size_t b  = gid >> 20;                  // N*N = 1<<20
  size_t qk = gid & ((1u << 20) - 1);
#pragma unroll
  for (int h = 0; h < Hn; ++h) {
    float acc = s_bz[h];
#pragma unroll
    for (int p = 0; p < Pn; ++p) acc += zv[p] * s_wz[h * Pn + p];
    out2[((b * Hn + h) << 20) + qk] = acc;  // RT store -> L2 resident for attn
  }
}

// ---------------------------------------------------------------------------
// Kernel 4: flash attention.  One wave owns (b, h, 16-query tile); streams
// 16-key tiles.  Scores: 8x WMMA over hd=32.  Online softmax with width-16
// shfl_xor row reductions.  P tile re-fragmented through LDS (wave-private
// 1 KB region) into A-layout, then P*V via 8x WMMA into 16x32 accumulators.
// qkv (12 MB) and proj_z (128 MB) are expected L2-resident here.
// ---------------------------------------------------------------------------
__global__ void attn_kernel(const float* __restrict__ qkv, const float* __restrict__ pzall,
                            const unsigned char* __restrict__ mask,
                            float* __restrict__ attn_out) {
  __shared__ float pst[8 * 256];
  int wave = threadIdx.x >> 5, lane = threadIdx.x & 31;
  float* ps = pst + wave * 256;

  int gt = blockIdx.x * 8 + wave;  // 2048 = B*H*(N/16)
  int b  = gt >> 9;
  int h  = (gt >> 6) & 7;
  int q0 = (gt & 63) * 16;

  int m     = lane & 15;
  int kh    = (lane >> 4) << 1;
  int rbase = (lane >> 4) << 3;

  const float* qb = qkv + (size_t)b * Nn * QKVW;

  // Q tile as 8 A-fragments (K chunks of 4 over hd=32)
  v2f qa[8];
#pragma unroll
  for (int c = 0; c < 8; ++c) {
    const float* p = qb + (size_t)(q0 + m) * QKVW + h * HDn + 4 * c + kh;
    qa[c].x = p[0];
    qa[c].y = p[1];
  }

  const float* pzb = pzall + (((size_t)(b * Hn + h) * Nn + q0) << 10);
  const unsigned char* mrow = mask + b * Nn;

  float rm[8], rl[8];
#pragma unroll
  for (int v = 0; v < 8; ++v) { rm[v] = -3.0e38f; rl[v] = 0.f; }
  v8f o0 = {}, o1 = {};
  const float scale = 0.17677669529663687f;  // 1/sqrt(32)

  for (int kt = 0; kt < 64; ++kt) {
    int k0 = kt * 16;
    // scores tile S = Q K^T (16x16)
    v8f sc = {};
#pragma unroll
    for (int c = 0; c < 8; ++c) {
      const float* p = qb + (size_t)(k0 + m) * QKVW + Dn + h * HDn + 4 * c + kh;
      v2f kb;
      kb.x = p[0];
      kb.y = p[1];
      sc = WMMA_F32_4(qa[c], kb, sc);
    }
    float mb = mrow[k0 + m] ? 0.0f : -1.0e9f;  // key mask bias (per column)
#pragma unroll
    for (int v = 0; v < 8; ++v) {
      int r = rbase + v;
      float sv = sc[v] * scale + pzb[((size_t)r << 10) + k0 + m] + mb;
      // row max over the 16 lanes of this half-wave
      float t = sv;
#pragma unroll
      for (int mk = 1; mk < 16; mk <<= 1) t = fmaxf(t, __shfl_xor(t, mk));
      float mnew = fmaxf(rm[v], t);
      float pr   = __expf(sv - mnew);
      float rsum = pr;
#pragma unroll
      for (int mk = 1; mk < 16; mk <<= 1) rsum += __shfl_xor(rsum, mk);
      float corr = __expf(rm[v] - mnew);
      rl[v] = rl[v] * corr + rsum;
      rm[v] = mnew;
      o0[v] *= corr;
      o1[v] *= corr;
      ps[r * 16 + m] = pr;  // stage P tile (row-major) for A-layout reload
    }
    // O += P * V   (P: 16x16, V: 16x32, two N-halves)
#pragma unroll
    for (int c2 = 0; c2 < 4; ++c2) {
      int kk = 4 * c2 + kh;
      v2f pa;
      pa.x = ps[m * 16 + kk];
      pa.y = ps[m * 16 + kk + 1];
      const float* vb = qb + (size_t)(k0 + kk) * QKVW + 2 * Dn + h * HDn;
      v2f v0, v1;
      v0.x = vb[m];
      v0.y = vb[QKVW + m];       // next key row, same d column
      v1.x = vb[16 + m];
      v1.y = vb[QKVW + 16 + m];
      o0 = WMMA_F32_4(pa, v0, o0);
      o1 = WMMA_F32_4(pa, v1, o1);
    }
  }
#pragma unroll
  for (int v = 0; v < 8; ++v) {
    int r = rbase + v;
    float inv = 1.0f / rl[v];
    float* orow = attn_out + (size_t)(b * Nn + q0 + r) * Dn + h * HDn;
    orow[m]      = o0[v] * inv;
    orow[16 + m] = o1[v] * inv;
  }
}

// ---------------------------------------------------------------------------
// Kernel 5: out = (attn_out * g) @ W_o^T + b_o + x   (residual), WMMA f32.
// Final output is written once and only read back by the host: NT stores.
// ---------------------------------------------------------------------------
__global__ void final_kernel(const float* __restrict__ ao, const float* __restrict__ g,
                             const float* __restrict__ wo, const float* __restrict__ bo,
                             const float* __restrict__ x, float* __restrict__ out1) {
  int wave = threadIdx.x >> 5, lane = threadIdx.x & 31;
  int tile = blockIdx.x * 8 + wave;  // 256 * 16 = 4096 tiles
  int mt = tile >> 4, nt = tile & 15;
  int m0 = mt * 16, n0 = nt * 16;
  int l16 = lane & 15, kh = (lane >> 4) << 1;
  const float* arow = ao + (size_t)(m0 + l16) * Dn;
  const float* grow = g + (size_t)(m0 + l16) * Dn;
  int n = n0 + l16;
  const float* brow = wo + (size_t)n * Dn;
  v8f c = {};
  for (int k = 0; k < Dn; k += 4) {
    v2f a, bv;
    a.x = arow[k + kh] * grow[k + kh];
    a.y = arow[k + kh + 1] * grow[k + kh + 1];
    bv.x = brow[k + kh];
    bv.y = brow[k + kh + 1];
    c = WMMA_F32_4(a, bv, c);
  }
  int rbase = (lane >> 4) << 3;
  float bias = bo[n];
#pragma unroll
  for (int v = 0; v < 8; ++v) {
    size_t idx = (size_t)(m0 + rbase + v) * Dn + n;
    __builtin_nontemporal_store(c[v] + bias + x[idx], out1 + idx);
  }
}

// ---------------------------------------------------------------------------
extern "C" void kernel_launch(void* const* d_in, const int* in_sizes, int n_in,
                              void* d_out, int out_size, void* d_ws, size_t ws_size,
                              hipStream_t stream) {
  const float* x        = (const float*)d_in[0];
  const float* z        = (const float*)d_in[1];
  const unsigned char* mask = (const unsigned char*)d_in[2];
  const float* qkv_w    = (const float*)d_in[3];
  const float* w_proj_z = (const float*)d_in[4];
  const float* w_proj_g = (const float*)d_in[5];
  const float* w_proj_o = (const float*)d_in[6];
  const float* w_ln_z   = (const float*)d_in[7];
  const float* b_ln_z   = (const float*)d_in[8];
  const float* b_proj_z = (const float*)d_in[9];
  const float* b_proj_g = (const float*)d_in[10];
  const float* b_proj_o = (const float*)d_in[11];
  const float* ln_w     = (const float*)d_in[12];
  const float* ln_b     = (const float*)d_in[13];

  float* out1 = (float*)d_out;                    // [B,N,D] residual output
  float* out2 = out1 + (size_t)Bn * Nn * Dn;      // [B,H,N,N] proj_z output

  float* ws   = (float*)d_ws;
  float* xn   = ws;                               // [B,N,D]     4 MB
  float* qkv  = xn + (size_t)Bn * Nn * Dn;        // [B,N,768]  12 MB
  float* gbuf = qkv + (size_t)Bn * Nn * 3 * Dn;   // [B,N,D]     4 MB
  float* aout = gbuf + (size_t)Bn * Nn * Dn;      // [B,N,D]     4 MB

  ln_x_kernel <<<512,   256, 0, stream>>>(x, ln_w, ln_b, xn);
  qkvg_kernel <<<2048,  256, 0, stream>>>(xn, qkv_w, w_proj_g, b_proj_g, qkv, gbuf);
  projz_kernel<<<16384, 256, 0, stream>>>(z, w_ln_z, b_ln_z, w_proj_z, b_proj_z, out2);
  attn_kernel <<<256,   256, 0, stream>>>(qkv, out2, mask, aout);
  final_kernel<<<512,   256, 0, stream>>>(aout, gbuf, w_proj_o, b_proj_o, x, out1);
}